// patch_extract_38860864094551
// MI455X (gfx1250) — compile-verified
//
#include <hip/hip_runtime.h>
#include <math.h>

// ---------------------------------------------------------------------------
// Problem constants (from reference): B=4096, L=512, F=20, CG=3, P=30, C=20
// ---------------------------------------------------------------------------
#define CH    20      // channels (F == C == 20)
#define POS   30      // patch length
#define ROFF  6       // left padding rows in LDS buffer (max pad = 6 for k=13)
#define ROWS  44      // logical rows -6 .. 37 (covers mt=1 tile reads up to row 31+12-6)
#define BUF   (ROWS * CH)   // 880 floats per buffer
#define EPSBN 1e-3f

typedef __attribute__((ext_vector_type(2))) float v2f;
typedef __attribute__((ext_vector_type(8))) float v8f;

struct BNp { const float *gamma, *beta, *mean, *var; };
struct RBp { const float *w1, *b1, *w2, *b2; BNp bn1, bn2; };
struct S2V { const float *c1w, *c1b; BNp bn1, bn2; RBp rb[4]; };

// ---------------------------------------------------------------------------
// WMMA conv1d layer:  Y[30x20] = sum_k X[shifted 30x20] * W_k[20x20] + bias
// One wave. X/Y in LDS with ROFF-row zero padding on both sides.
// A frag (16x4 f32): lane m = lane&15; VGPR0/1 = K {0,1} (lanes 0-15) / {2,3} (16-31)
// B frag (4x16 f32): lane n = lane&15; same K split by lane half.
// D (16x16 f32, 8 VGPR): M = r + (half?8:0), N = lane&15.
// ---------------------------------------------------------------------------
template <int K, int DIL>
__device__ inline void conv_wmma(const float* __restrict__ w,
                                 const float* __restrict__ bias,
                                 const float* __restrict__ xin,
                                 float* __restrict__ yout) {
  const int lane = (int)(threadIdx.x & 31u);
  const int m    = lane & 15;
  const int hl   = lane >> 4;               // 0 or 1
  constexpr int PAD = ((K - 1) * DIL) / 2;  // keras SAME pad_left

  for (int nt = 0; nt < 2; ++nt) {
    const int   co    = nt * 16 + m;                 // output channel this lane covers
    const int   coc   = (co < CH) ? co : 0;          // clamped (keep loads in-bounds)
    const float wmask = (co < CH) ? 1.0f : 0.0f;     // zero out padded N columns
    v8f acc0 = {};   // M rows 0..15
    v8f acc1 = {};   // M rows 16..31 (30,31 discarded)
    for (int k = 0; k < K; ++k) {
      const float* xr0 = xin + (m + k * DIL - PAD + ROFF) * CH;
      const float* xr1 = xr0 + 16 * CH;
      const float* wk  = w + k * CH * CH;            // W[k][ci][co], row-major
#pragma unroll
      for (int kc = 0; kc < 5; ++kc) {               // Cin = 20 = 5 chunks of 4
        const int ci = kc * 4 + hl * 2;
        const float* wp = wk + ci * CH + coc;
        v2f bb; bb.x = wp[0] * wmask; bb.y = wp[CH] * wmask;
        v2f a0; a0.x = xr0[ci]; a0.y = xr0[ci + 1];
        v2f a1; a1.x = xr1[ci]; a1.y = xr1[ci + 1];
        acc0 = __builtin_amdgcn_wmma_f32_16x16x4_f32(false, a0, false, bb,
                                                     (short)0, acc0, false, false);
        acc1 = __builtin_amdgcn_wmma_f32_16x16x4_f32(false, a1, false, bb,
                                                     (short)0, acc1, false, false);
      }
    }
    if (co < CH) {
      const float bv = bias[co];
#pragma unroll
      for (int r = 0; r < 8; ++r) {
        const int m0 = r + hl * 8;                   // rows 0..15: always valid
        yout[(m0 + ROFF) * CH + co] = acc0[r] + bv;
        const int m1 = m0 + 16;                      // rows 16..31: clip at 30
        if (m1 < POS) yout[(m1 + ROFF) * CH + co] = acc1[r] + bv;
      }
    }
  }
}

// ---------------------------------------------------------------------------
// Pointwise epilogues (per-lane over 600 elements)
// ---------------------------------------------------------------------------
__device__ inline void pw_elu(float* buf) {
  for (int i = (int)threadIdx.x; i < POS * CH; i += 32) {
    const int r = i / CH, c = i - r * CH;
    const int idx = (r + ROFF) * CH + c;
    const float x = buf[idx];
    buf[idx] = (x > 0.0f) ? x : (__expf(x) - 1.0f);
  }
}

__device__ inline void pw_bn(float* buf, const BNp& bn, bool relu) {
  for (int i = (int)threadIdx.x; i < POS * CH; i += 32) {
    const int r = i / CH, c = i - r * CH;
    const int idx = (r + ROFF) * CH + c;
    const float x = buf[idx];
    float y = (x - bn.mean[c]) * (bn.gamma[c] * rsqrtf(bn.var[c] + EPSBN)) + bn.beta[c];
    if (relu) y = fmaxf(y, 0.0f);
    buf[idx] = y;
  }
}

__device__ inline void pw_bn_add_relu(float* U, const BNp& bn, const float* X) {
  for (int i = (int)threadIdx.x; i < POS * CH; i += 32) {
    const int r = i / CH, c = i - r * CH;
    const int idx = (r + ROFF) * CH + c;
    const float x = U[idx];
    float y = (x - bn.mean[c]) * (bn.gamma[c] * rsqrtf(bn.var[c] + EPSBN)) + bn.beta[c];
    U[idx] = fmaxf(y + X[idx], 0.0f);
  }
}

template <int K>
__device__ inline void resblock(const RBp& rb, float*& X, float*& T, float*& U) {
  conv_wmma<K, 1>(rb.w1, rb.b1, X, T);  __syncthreads();
  pw_bn(T, rb.bn1, true);               __syncthreads();
  conv_wmma<K, 2>(rb.w2, rb.b2, T, U);  __syncthreads();   // dilation 2 second conv
  pw_bn_add_relu(U, rb.bn2, X);         __syncthreads();
  float* tmp = X; X = U; U = tmp;
}

// ---------------------------------------------------------------------------
// Kernel 1: one wave per batch element; full seq2vec network in LDS.
// ---------------------------------------------------------------------------
__global__ __launch_bounds__(32) void seq2vec_kernel(const float* __restrict__ seq,
                                                     const int* __restrict__ patch,
                                                     int s, S2V P,
                                                     float* __restrict__ vout, int Lseq) {
  __shared__ float lds[3 * BUF];
  float* bufA = lds;
  float* bufB = lds + BUF;
  float* bufC = lds + 2 * BUF;
  const int b = (int)blockIdx.x;

  for (int i = (int)threadIdx.x; i < 3 * BUF; i += 32) lds[i] = 0.0f;   // zero pads
  __syncthreads();

  const int off = patch[b * 2 + s];
  for (int i = (int)threadIdx.x; i < POS * CH; i += 32) {
    const int r = i / CH, c = i - r * CH;
    bufA[(r + ROFF) * CH + c] = seq[((size_t)b * Lseq + off + r) * CH + c];
  }
  __syncthreads();

  conv_wmma<13, 1>(P.c1w, P.c1b, bufA, bufB);  __syncthreads();
  pw_elu(bufB);                                __syncthreads();
  pw_bn(bufB, P.bn1, true);                    __syncthreads();

  float* X = bufB; float* T = bufC; float* U = bufA;
  resblock<7>(P.rb[0], X, T, U);
  resblock<5>(P.rb[1], X, T, U);
  pw_bn(X, P.bn2, false);  __syncthreads();
  resblock<5>(P.rb[2], X, T, U);
  pw_bn(X, P.bn1, false);  __syncthreads();     // bn1 reused, as in the source
  resblock<5>(P.rb[3], X, T, U);

  float* vd = vout + (size_t)b * (POS * CH);
  for (int i = (int)threadIdx.x; i < POS * CH; i += 32) {
    const int r = i / CH, c = i - r * CH;
    vd[i] = X[(r + ROFF) * CH + c];
  }
}

// ---------------------------------------------------------------------------
// Kernel 2: bandwidth-bound broadcast + concat -> out[b, i, j, 43]
// ---------------------------------------------------------------------------
__global__ __launch_bounds__(256) void assemble_kernel(const float* __restrict__ geo,
                                                       const float* __restrict__ v,
                                                       float* __restrict__ out, int nB) {
  const int b = (int)blockIdx.x;
  __shared__ float s1[POS * CH];
  __shared__ float s2[POS * CH];
  const float* v1 = v + (size_t)b * (POS * CH);
  const float* v2 = v + ((size_t)nB + b) * (POS * CH);
  for (int i = (int)threadIdx.x; i < POS * CH; i += 256) { s1[i] = v1[i]; s2[i] = v2[i]; }
  __syncthreads();

  const float* g = geo + (size_t)b * (POS * POS * 3);
  float* o = out + (size_t)b * (POS * POS * 43);
  for (int idx = (int)threadIdx.x; idx < POS * POS * 43; idx += 256) {
    const int c  = idx % 43;
    const int ij = idx / 43;
    const int i2 = ij / POS;
    const int j2 = ij - i2 * POS;
    float val;
    if (c < 3)       val = g[ij * 3 + c];
    else if (c < 23) val = s1[i2 * CH + (c - 3)];
    else             val = s2[j2 * CH + (c - 23)];
    o[idx] = val;
  }
}

// ---------------------------------------------------------------------------
// Host: detect flattened input ordering and build param pointer structs.
// ---------------------------------------------------------------------------
static void fill_bn_sorted(BNp& bn, void* const* d_in, int base) {  // beta,gamma,mean,var
  bn.beta  = (const float*)d_in[base + 0];
  bn.gamma = (const float*)d_in[base + 1];
  bn.mean  = (const float*)d_in[base + 2];
  bn.var   = (const float*)d_in[base + 3];
}
static void fill_bn_ins(BNp& bn, void* const* d_in, int base) {     // gamma,beta,mean,var
  bn.gamma = (const float*)d_in[base + 0];
  bn.beta  = (const float*)d_in[base + 1];
  bn.mean  = (const float*)d_in[base + 2];
  bn.var   = (const float*)d_in[base + 3];
}

static void fill_s2v(S2V& p, void* const* d_in, int o, bool sorted_nested) {
  if (sorted_nested) {
    // sorted keys: bn1, bn2, c1b, c1w, rb1..rb4 (rb: b1,b2,bn1,bn2,w1,w2)
    fill_bn_sorted(p.bn1, d_in, o + 0);
    fill_bn_sorted(p.bn2, d_in, o + 4);
    p.c1b = (const float*)d_in[o + 8];
    p.c1w = (const float*)d_in[o + 9];
    for (int i = 0; i < 4; ++i) {
      const int r = o + 10 + 12 * i;
      p.rb[i].b1 = (const float*)d_in[r + 0];
      p.rb[i].b2 = (const float*)d_in[r + 1];
      fill_bn_sorted(p.rb[i].bn1, d_in, r + 2);
      fill_bn_sorted(p.rb[i].bn2, d_in, r + 6);
      p.rb[i].w1 = (const float*)d_in[r + 10];
      p.rb[i].w2 = (const float*)d_in[r + 11];
    }
  } else {
    // insertion order: c1w, c1b, bn1, bn2, rb1..rb4 (rb: w1,b1,w2,b2,bn1,bn2)
    p.c1w = (const float*)d_in[o + 0];
    p.c1b = (const float*)d_in[o + 1];
    fill_bn_ins(p.bn1, d_in, o + 2);
    fill_bn_ins(p.bn2, d_in, o + 6);
    for (int i = 0; i < 4; ++i) {
      const int r = o + 10 + 12 * i;
      p.rb[i].w1 = (const float*)d_in[r + 0];
      p.rb[i].b1 = (const float*)d_in[r + 1];
      p.rb[i].w2 = (const float*)d_in[r + 2];
      p.rb[i].b2 = (const float*)d_in[r + 3];
      fill_bn_ins(p.rb[i].bn1, d_in, r + 4);
      fill_bn_ins(p.rb[i].bn2, d_in, r + 8);
    }
  }
}

extern "C" void kernel_launch(void* const* d_in, const int* in_sizes, int n_in,
                              void* d_out, int out_size, void* d_ws, size_t ws_size,
                              hipStream_t stream) {
  (void)out_size; (void)ws_size; (void)n_in;

  const int nB = in_sizes[0] / (POS * POS * 3);   // geo_patch is leaf 0 in both orders

  int idx_patch, idx_seq1, idx_seq2, p1o, p2o;
  if (in_sizes[1] == 2 * nB) {
    // top-level insertion order: geo, patch, seq1M, seq2M, params1, params2
    idx_patch = 1; idx_seq1 = 2; idx_seq2 = 3; p1o = 4; p2o = 4 + 58;
  } else {
    // top-level sorted order: geo, params1(58), params2(58), patch, seq1M, seq2M
    p1o = 1; p2o = 59; idx_patch = 117; idx_seq1 = 118; idx_seq2 = 119;
  }
  const bool sorted_nested = (in_sizes[p1o] < 100);   // bn vector (20) vs c1w (5200)

  const int Lseq = in_sizes[idx_seq1] / (nB * CH);

  S2V P1, P2;
  fill_s2v(P1, d_in, p1o, sorted_nested);
  fill_s2v(P2, d_in, p2o, sorted_nested);

  float* v = (float*)d_ws;   // [2][nB][30][20] floats

  seq2vec_kernel<<<nB, 32, 0, stream>>>((const float*)d_in[idx_seq1],
                                        (const int*)d_in[idx_patch], 0, P1, v, Lseq);
  seq2vec_kernel<<<nB, 32, 0, stream>>>((const float*)d_in[idx_seq2],
                                        (const int*)d_in[idx_patch], 1, P2,
                                        v + (size_t)nB * (POS * CH), Lseq);
  assemble_kernel<<<nB, 256, 0, stream>>>((const float*)d_in[0], v, (float*)d_out, nB);
}